// FixedLengthDecoder_6150393168259
// MI455X (gfx1250) — compile-verified
//
#include <hip/hip_runtime.h>
#include <math.h>

typedef float v2f __attribute__((ext_vector_type(2)));
typedef float v8f __attribute__((ext_vector_type(8)));

__device__ __forceinline__ v8f wmma_f32x4(v2f a, v2f b, v8f c) {
  // D = A(16x4, f32) x B(4x16, f32) + C(16x16, f32)   [wave32]
  return __builtin_amdgcn_wmma_f32_16x16x4_f32(false, a, false, b, (short)0, c,
                                               false, false);
}

// C[M x N] = act( A1[M x K1] @ W1[K1 x N] + A2[M x K2] @ W2[K2 x N] + bias )
// One wave -> 16x32 output strip (2 WMMA tiles, shared A fragment).
// 8 waves per block share the same 16 M-rows; block covers 16 x 256.
__global__ __launch_bounds__(256) void gemm_dual_wmma(
    const float* __restrict__ A1, long lda1, const float* __restrict__ W1, int K1,
    const float* __restrict__ A2, long lda2, const float* __restrict__ W2, int K2,
    int ldw, const float* __restrict__ bias,
    float* __restrict__ C, long ldc, int act)
{
  const int lane = threadIdx.x & 31;
  const int wave = threadIdx.x >> 5;
  const int hi   = lane >> 4;        // 0: lanes 0-15, 1: lanes 16-31
  const int l16  = lane & 15;
  const int mbase = blockIdx.y * 16;
  const int n0    = blockIdx.x * 256 + wave * 32 + l16;

  v8f acc0 = {0.f,0.f,0.f,0.f,0.f,0.f,0.f,0.f};
  v8f acc1 = acc0;

  for (int p = 0; p < 2; ++p) {
    const float* A  = p ? A2 : A1;
    const float* W  = p ? W2 : W1;
    const long  lda = p ? lda2 : lda1;
    const int   K   = p ? K2 : K1;
    if (K == 0) continue;                       // uniform branch (scalar arg)
    const float* arow = A + (long)(mbase + l16) * lda;
    #pragma unroll 4
    for (int k0 = 0; k0 < K; k0 += 4) {
      const int ka = k0 + 2 * hi;
      v2f a; a.x = arow[ka]; a.y = arow[ka + 1];
      const float* wr0 = W + (long)ka * ldw;
      const float* wr1 = wr0 + ldw;
      v2f b0; b0.x = wr0[n0];      b0.y = wr1[n0];
      v2f b1; b1.x = wr0[n0 + 16]; b1.y = wr1[n0 + 16];
      acc0 = wmma_f32x4(a, b0, acc0);
      acc1 = wmma_f32x4(a, b1, acc1);
    }
  }

  #pragma unroll
  for (int v = 0; v < 8; ++v) {
    const int row = mbase + v + 8 * hi;
    float x0 = acc0[v];
    float x1 = acc1[v];
    if (bias) { x0 += bias[n0]; x1 += bias[n0 + 16]; }
    if (act)  { x0 = tanhf(x0); x1 = tanhf(x1); }
    C[(long)row * ldc + n0]      = x0;
    C[(long)row * ldc + n0 + 16] = x1;
  }
}

// One block (256 threads) per batch row b:
//   LSTM gates -> h_new/c_new, Luong attention (score/softmax/weighted).
// Writes h_new to hw[b][0:512], weighted to hw[b][512:1024], c in place.
__global__ __launch_bounds__(256) void gates_attn(
    const float* __restrict__ z, float* __restrict__ c,
    float* __restrict__ hw, const float* __restrict__ src)
{
  constexpr int S = 256, U = 512;
  const int b   = blockIdx.x;
  const int tid = threadIdx.x;
  __shared__ float h_s[U];
  __shared__ float a_s[S];
  __shared__ float red[256];

  // ---- gates (keras order i,f,g,o over 4U columns of z) ----
  const float* zb = z + (long)b * 4 * U;
  for (int u = tid; u < U; u += 256) {
    const float zi = zb[u];
    const float zf = zb[u + U];
    const float zg = zb[u + 2 * U];
    const float zo = zb[u + 3 * U];
    const float ig = 1.f / (1.f + expf(-zi));
    const float fg = 1.f / (1.f + expf(-zf));
    const float gg = tanhf(zg);
    const float og = 1.f / (1.f + expf(-zo));
    const float cn = fg * c[(long)b * U + u] + ig * gg;
    const float hn = og * tanhf(cn);
    c[(long)b * U + u]    = cn;
    hw[(long)b * 2 * U + u] = hn;
    h_s[u] = hn;
  }
  __syncthreads();

  // ---- score[s] = src[b,s,:] . h_new  (thread tid handles s = tid) ----
  const float4* row = (const float4*)(src + ((long)b * S + tid) * U);
  float sc = 0.f;
  #pragma unroll 4
  for (int q = 0; q < U / 4; ++q) {
    const float4 v = row[q];
    sc += v.x * h_s[4 * q] + v.y * h_s[4 * q + 1] +
          v.z * h_s[4 * q + 2] + v.w * h_s[4 * q + 3];
  }

  // ---- softmax over S=256 (block reductions) ----
  red[tid] = sc; __syncthreads();
  for (int off = 128; off; off >>= 1) {
    if (tid < off) red[tid] = fmaxf(red[tid], red[tid + off]);
    __syncthreads();
  }
  const float mx = red[0]; __syncthreads();
  const float e = expf(sc - mx);
  red[tid] = e; __syncthreads();
  for (int off = 128; off; off >>= 1) {
    if (tid < off) red[tid] += red[tid + off];
    __syncthreads();
  }
  a_s[tid] = e * (1.f / red[0]);
  __syncthreads();

  // ---- weighted[u] = sum_s a_s[s] * src[b,s,u]  (coalesced over u) ----
  for (int u = tid; u < U; u += 256) {
    const float* col = src + (long)b * S * U + u;
    float wsum = 0.f;
    #pragma unroll 8
    for (int s = 0; s < S; ++s) wsum += a_s[s] * col[(long)s * U];
    hw[(long)b * 2 * U + U + u] = wsum;
  }
}

// Copy h0 into the strided concat buffer and c0 into scratch (inputs immutable).
__global__ __launch_bounds__(256) void init_hc(
    const float* __restrict__ h0, const float* __restrict__ c0,
    float* __restrict__ hw, float* __restrict__ c)
{
  const int i = blockIdx.x * blockDim.x + threadIdx.x;   // [0, 128*512)
  const int b = i >> 9, u = i & 511;
  hw[(long)b * 1024 + u] = h0[i];
  c[i] = c0[i];
}

extern "C" void kernel_launch(void* const* d_in, const int* in_sizes, int n_in,
                              void* d_out, int out_size, void* d_ws, size_t ws_size,
                              hipStream_t stream) {
  (void)in_sizes; (void)n_in; (void)out_size; (void)ws_size;
  const float* h0   = (const float*)d_in[0];
  const float* c0   = (const float*)d_in[1];
  const float* src  = (const float*)d_in[2];   // [B,S,U]
  const float* tgt  = (const float*)d_in[3];   // [B,T+1,U]
  const float* kern = (const float*)d_in[4];   // [U,4U]
  const float* rker = (const float*)d_in[5];   // [U,4U]
  const float* bias = (const float*)d_in[6];   // [4U]
  const float* attw = (const float*)d_in[7];   // [2U,U]

  constexpr int B = 128, S = 256, T = 256, U = 512;
  (void)S;
  float* z   = (float*)d_ws;             // [B, 4U]
  float* hw  = z  + (long)B * 4 * U;     // [B, 2U]  = concat(h_new, weighted)
  float* c   = hw + (long)B * 2 * U;     // [B, U]
  float* out = (float*)d_out;            // [B, T, U]

  init_hc<<<(B * U) / 256, 256, 0, stream>>>(h0, c0, hw, c);

  for (int t = 0; t < T; ++t) {
    const float* xt = tgt + (long)t * U;          // lda = (T+1)*U
    // z = x_t @ kernel + h @ recurrent_kernel + bias   (M=128,N=2048,K=512+512)
    gemm_dual_wmma<<<dim3(8, B / 16), 256, 0, stream>>>(
        xt, (long)(T + 1) * U, kern, U,
        hw, (long)2 * U,       rker, U,
        4 * U, bias, z, (long)4 * U, /*act=*/0);

    gates_attn<<<B, 256, 0, stream>>>(z, c, hw, src);

    // out[:,t,:] = tanh( concat(h,weighted) @ attn_w )  (M=128,N=512,K=1024)
    gemm_dual_wmma<<<dim3(2, B / 16), 256, 0, stream>>>(
        hw, (long)2 * U, attw, 2 * U,
        nullptr, 0, nullptr, 0,
        U, nullptr, out + (long)t * U, (long)T * U, /*act=*/1);
  }
}